// NOENet_32229434589285
// MI455X (gfx1250) — compile-verified
//
#include <hip/hip_runtime.h>

// CDNA5 / gfx1250, wave32. Uses V_WMMA_F32_16X16X4_F32 (16x4 A, 4x16 B, 16x16 C/D).
// Transposed formulation: z.T(32x16) = W1(32x4) x inp.T(4x16), hidden on M, batch on N.
// -log2(e) is pre-folded into W1/b1 so WMMA output feeds v_exp_f32 directly.
typedef float v2f __attribute__((ext_vector_type(2)));
typedef float v8f __attribute__((ext_vector_type(8)));

#define HIDDEN 32
#define NL2E (-1.44269504088896340736f)   // -log2(e)

__device__ __forceinline__ v8f wmma4(v2f a, v2f b, v8f c) {
    // D = A(16x4) x B(4x16) + C ; 8 args: neg_a, A, neg_b, B, c_mod, C, reuse_a, reuse_b
    return __builtin_amdgcn_wmma_f32_16x16x4_f32(false, a, false, b, (short)0, c,
                                                 false, false);
}

__device__ __forceinline__ v8f sigmoid8_pre(v8f x) {
    // x already equals -log2(e)*(Wu+b): sigmoid = rcp(1 + exp2(x))
    v8f r;
#pragma unroll
    for (int i = 0; i < 8; ++i) {
        float e = __builtin_amdgcn_exp2f(x[i]);
        r[i] = __builtin_amdgcn_rcpf(1.0f + e);
    }
    return r;
}

__global__ __launch_bounds__(256) void noenet_scan_wmma(
    const float* __restrict__ x,
    const float* __restrict__ l1w, const float* __restrict__ l1b,
    const float* __restrict__ l2w, const float* __restrict__ l2b,
    const float* __restrict__ r1w, const float* __restrict__ r1b,
    const float* __restrict__ r2w, const float* __restrict__ r2b,
    float* __restrict__ out, int B, int T)
{
    const int lane = threadIdx.x & 31;
    const int wave = (blockIdx.x * blockDim.x + threadIdx.x) >> 5;
    const int wb = wave * 16;                 // 16 batch elements per wave
    if (wb >= B) return;

    const int lm = lane & 15;                 // batch-in-tile index / weight row
    const int hl = lane >> 4;                 // half-wave select
    const int k0 = hl ? 2 : 0;                // A/B K striping: VGPR0 holds K=0|2, VGPR1 K=1|3

    // ---- A matrices: (-log2e * W1) / (-log2e * rW1) as 16x4 tiles, resident in VGPRs
    v2f a1_lo, a1_hi, ar_lo, ar_hi;           // lo: hidden 0-15 (M), hi: hidden 16-31
    a1_lo[0] = NL2E * l1w[lm * 4 + k0];        a1_lo[1] = NL2E * l1w[lm * 4 + k0 + 1];
    a1_hi[0] = NL2E * l1w[(lm + 16) * 4 + k0]; a1_hi[1] = NL2E * l1w[(lm + 16) * 4 + k0 + 1];
    ar_lo[0] = NL2E * r1w[lm * 4 + k0];        ar_lo[1] = NL2E * r1w[lm * 4 + k0 + 1];
    ar_hi[0] = NL2E * r1w[(lm + 16) * 4 + k0]; ar_hi[1] = NL2E * r1w[(lm + 16) * 4 + k0 + 1];

    // ---- Bias as C matrices (also pre-scaled): D layout M = r + 8*hl, N = lane%16
    v8f c1_lo, c1_hi, cr_lo, cr_hi;
#pragma unroll
    for (int r = 0; r < 8; ++r) {
        const int m = r + 8 * hl;
        c1_lo[r] = NL2E * l1b[m];        c1_hi[r] = NL2E * l1b[16 + m];
        cr_lo[r] = NL2E * r1b[m];        cr_hi[r] = NL2E * r1b[16 + m];
    }

    // ---- Layer-2 weights, per-lane slices matching the D layout (readout = in-lane dot)
    float w2A[8], w2B[8], rwA0[8], rwB0[8], rwA1[8], rwB1[8];
#pragma unroll
    for (int r = 0; r < 8; ++r) {
        const int iA = r + 8 * hl, iB = iA + 16;
        w2A[r]  = l2w[iA];            w2B[r]  = l2w[iB];
        rwA0[r] = r2w[iA];            rwB0[r] = r2w[iB];
        rwA1[r] = r2w[HIDDEN + iA];   rwB1[r] = r2w[HIDDEN + iB];
    }
    const float b2 = l2b[0], rb0 = r2b[0], rb1 = r2b[1];

    float h0 = 0.0f, h1 = 0.0f;               // recurrent state for batch lm (all lanes)

    const int TO = T - 1;                     // number of scan steps
    const float* xrow = x + (size_t)(wb + lm) * (size_t)T;
    float* orow = out + (size_t)(wb + lm) * (size_t)TO;

    // one scan step for 16 batch elements; returns the readout for batch lm
    auto step = [&](float u0, float u1) -> float {
        v2f b;                                // B matrix 4x16: K=0|2 in b[0], K=1|3 in b[1]
        b[0] = hl ? h0 : u0;                  // lanes 0-15: u ; lanes 16-31: h
        b[1] = hl ? h1 : u1;
        v8f z_lo = wmma4(a1_lo, b, c1_lo);
        v8f z_hi = wmma4(a1_hi, b, c1_hi);
        v8f g_lo = wmma4(ar_lo, b, cr_lo);
        v8f g_hi = wmma4(ar_hi, b, cr_hi);
        z_lo = sigmoid8_pre(z_lo); z_hi = sigmoid8_pre(z_hi);
        g_lo = sigmoid8_pre(g_lo); g_hi = sigmoid8_pre(g_hi);
        float po = 0.0f, p0 = 0.0f, p1 = 0.0f;
#pragma unroll
        for (int r = 0; r < 8; ++r) {
            po = fmaf(z_lo[r], w2A[r],  fmaf(z_hi[r], w2B[r],  po));
            p0 = fmaf(g_lo[r], rwA0[r], fmaf(g_hi[r], rwB0[r], p0));
            p1 = fmaf(g_lo[r], rwA1[r], fmaf(g_hi[r], rwB1[r], p1));
        }
        // halves hold disjoint hidden ranges -> single cross-half add completes the dot
        po += __shfl_xor(po, 16, 32);
        p0 += __shfl_xor(p0, 16, 32);
        p1 += __shfl_xor(p1, 16, 32);
        h0 = p0 + rb0;                        // lands in exactly the lanes next B needs
        h1 = p1 + rb1;
        return po + b2;
    };

    // x stream: aligned float4 chunks, branchless clamped prefetch one chunk ahead
    float4 xa = *(const float4*)(xrow);                            // cols c..c+3
    float4 xb = (T > 4) ? *(const float4*)(xrow + 4)               // cols c+4..c+7
                        : make_float4(0.f, 0.f, 0.f, 0.f);

    const int mainEnd = TO & ~3;              // full 4-step chunks, no guards inside
    int c = 0;
    for (; c < mainEnd; c += 4) {
        int pc = c + 8;                       // clamp: tail re-reads valid data, never used
        if (pc > T - 4) pc = T - 4;
        float4 xc = *(const float4*)(xrow + pc);

        float o0 = step(xa.x, xa.y);
        float o1 = step(xa.y, xa.z);
        float o2 = step(xa.z, xa.w);
        float o3 = step(xa.w, xb.x);

        if (hl == 0) {                        // halves are duplicates; store from lanes 0-15
            orow[c]     = o0;
            orow[c + 1] = o1;
            orow[c + 2] = o2;
            orow[c + 3] = o3;
        }
        xa = xb; xb = xc;
    }

    // tail: 0-3 remaining steps, all columns within xa
    if (c < TO) {
        float o0 = step(xa.x, xa.y);
        if (hl == 0) orow[c] = o0;
        if (c + 1 < TO) {
            float o1 = step(xa.y, xa.z);
            if (hl == 0) orow[c + 1] = o1;
        }
        if (c + 2 < TO) {
            float o2 = step(xa.z, xa.w);
            if (hl == 0) orow[c + 2] = o2;
        }
    }
}

extern "C" void kernel_launch(void* const* d_in, const int* in_sizes, int n_in,
                              void* d_out, int out_size, void* d_ws, size_t ws_size,
                              hipStream_t stream) {
    const float* x   = (const float*)d_in[0];
    const float* l1w = (const float*)d_in[1];
    const float* l1b = (const float*)d_in[2];
    const float* l2w = (const float*)d_in[3];
    const float* l2b = (const float*)d_in[4];
    const float* r1w = (const float*)d_in[5];
    const float* r1b = (const float*)d_in[6];
    const float* r2w = (const float*)d_in[7];
    const float* r2b = (const float*)d_in[8];
    float* out = (float*)d_out;

    const int n = in_sizes[0];                 // B*T
    const int B = n - out_size;                // B*T - B*(T-1) = B
    const int T = n / B;

    const int nwaves  = (B + 15) / 16;
    const int threads = 256;                   // 8 wave32 per workgroup
    const int grid    = (nwaves * 32 + threads - 1) / threads;

    noenet_scan_wmma<<<grid, threads, 0, stream>>>(
        x, l1w, l1b, l2w, l2b, r1w, r1b, r2w, r2b, out, B, T);
}